// diffECELoss_47923245089176
// MI455X (gfx1250) — compile-verified
//
#include <hip/hip_runtime.h>
#include <hip/hip_bf16.h>
#include <stdint.h>

#define N_BINS    15
#define CCOUNT    50      // classes per row (fixed by the reference problem)
#define TILE_ROWS 256     // rows staged in LDS per tile (256*50*4B = 50 KB)
#define BLOCK     256     // 8 waves of 32 on CDNA5
#define MAX_GRID  2048    // persistent blocks; keeps global-atomic traffic tiny

// ---------------------------------------------------------------------------
// CDNA5 async global->LDS copy path (GLOBAL_LOAD_ASYNC_TO_LDS_B128, ASYNCcnt)
// ---------------------------------------------------------------------------
#if defined(__gfx1250__) && __has_builtin(__builtin_amdgcn_global_load_async_to_lds_b128)
#define USE_ASYNC_LDS 1
#endif

#ifdef USE_ASYNC_LDS
// The builtin takes typed pointers: v4i32 in addrspace(1) (global) and
// addrspace(3) (LDS). C++ forbids implicit void* -> T* conversion, so build
// the exact types.
typedef int v4i __attribute__((ext_vector_type(4)));
typedef __attribute__((address_space(1))) v4i gv4i_t;
typedef __attribute__((address_space(3))) v4i lv4i_t;

// Generic shared pointers carry the LDS byte offset in their low 32 bits on
// AMDGPU; reinterpret through an integer to get an addrspace(3) pointer.
__device__ __forceinline__ lv4i_t* to_lds(void* p) {
  return reinterpret_cast<lv4i_t*>(
      static_cast<uint32_t>(reinterpret_cast<uintptr_t>(p)));
}
__device__ __forceinline__ gv4i_t* to_global(const void* p) {
  return reinterpret_cast<gv4i_t*>(reinterpret_cast<uintptr_t>(p));
}
#endif

__device__ __forceinline__ void wait_async_zero() {
#ifdef USE_ASYNC_LDS
#if __has_builtin(__builtin_amdgcn_s_wait_asynccnt)
  __builtin_amdgcn_s_wait_asynccnt(0);
#else
  asm volatile("s_wait_asynccnt 0" ::: "memory");
#endif
#endif
}

// ---------------------------------------------------------------------------
// Kernel 1: per-row confidence/accuracy, 15-bin histogram partials in d_ws.
// partials layout: [0..14]=counts, [15..29]=sum_acc, [30..44]=sum_conf
// ---------------------------------------------------------------------------
__global__ __launch_bounds__(BLOCK) void ece_partials_kernel(
    const float* __restrict__ logits, const long long* __restrict__ labels,
    float* __restrict__ partials, int nRows, int nTiles) {
  __shared__ float tile[TILE_ROWS * CCOUNT];
  __shared__ float s_cnt[N_BINS];
  __shared__ float s_acc[N_BINS];
  __shared__ float s_conf[N_BINS];

  const int tid = threadIdx.x;
  if (tid < N_BINS) { s_cnt[tid] = 0.f; s_acc[tid] = 0.f; s_conf[tid] = 0.f; }
  __syncthreads();

  for (int t = blockIdx.x; t < nTiles; t += gridDim.x) {
    const int rowBase  = t * TILE_ROWS;
    const int rowsHere = min(TILE_ROWS, nRows - rowBase);
    const int totalF   = rowsHere * CCOUNT;
    const int nChunks  = totalF >> 2;  // float4 chunks (tile base is 16B-aligned)
    const float* gsrc  = logits + (size_t)rowBase * (size_t)CCOUNT;

#ifdef USE_ASYNC_LDS
    // Coalesced streaming copy: 16 bytes/lane, global -> LDS, no VGPR staging.
    for (int idx = tid; idx < nChunks; idx += BLOCK) {
      __builtin_amdgcn_global_load_async_to_lds_b128(
          to_global(gsrc + idx * 4), to_lds(&tile[idx * 4]),
          /*offset=*/0, /*cpol=*/0);
    }
    for (int idx = (nChunks << 2) + tid; idx < totalF; idx += BLOCK)
      tile[idx] = gsrc[idx];               // scalar tail (empty for N%256==0)
    wait_async_zero();                     // ASYNCcnt -> 0 for this wave
#else
    const float4* g4 = (const float4*)gsrc;
    float4*       t4 = (float4*)tile;
    for (int idx = tid; idx < nChunks; idx += BLOCK) t4[idx] = g4[idx];
    for (int idx = (nChunks << 2) + tid; idx < totalF; idx += BLOCK)
      tile[idx] = gsrc[idx];
#endif
    __syncthreads();

    if (tid < rowsHere) {
      // Row stride 50 floats: lane addresses 50*t mod 64 hit 32 distinct
      // banks within a wave32 -> conflict-free LDS reads.
      const float* row = &tile[tid * CCOUNT];
      float m  = row[0];
      int   arg = 0;
#pragma unroll
      for (int k = 1; k < CCOUNT; ++k) {
        float v = row[k];
        if (v > m) { m = v; arg = k; }     // first-max wins, matches argmax
      }
      float s = 0.f;
#pragma unroll
      for (int k = 0; k < CCOUNT; ++k) s += __expf(row[k] - m);
      const float conf = 1.0f / s;         // == max(softmax(row))
      const int   lab  = (int)labels[rowBase + tid];
      const float acc  = (arg == lab) ? 1.0f : 0.0f;

      // bin i covers (i/15, (i+1)/15]; searchsorted-left on uppers ==
      // ceil(15*conf)-1, clamped.
      int bin = (int)ceilf(conf * (float)N_BINS) - 1;
      bin = bin < 0 ? 0 : (bin > N_BINS - 1 ? N_BINS - 1 : bin);

      atomicAdd(&s_cnt[bin], 1.0f);        // ds_add_f32
      atomicAdd(&s_acc[bin], acc);
      atomicAdd(&s_conf[bin], conf);
    }
    __syncthreads();                       // tile reused next iteration
  }

  if (tid < N_BINS) {                      // 45 global atomics per block
    atomicAdd(&partials[tid],              s_cnt[tid]);
    atomicAdd(&partials[N_BINS + tid],     s_acc[tid]);
    atomicAdd(&partials[2 * N_BINS + tid], s_conf[tid]);
  }
}

// ---------------------------------------------------------------------------
// Kernel 2: finalize 45 partials -> out[31] = ece(1) ++ over_conf(15) ++ prop(15)
// ---------------------------------------------------------------------------
__global__ void ece_finalize_kernel(const float* __restrict__ partials,
                                    float* __restrict__ out, float invN) {
  __shared__ float terms[N_BINS];
  const int i = threadIdx.x;
  if (i < N_BINS) {
    const float cnt  = partials[i];
    const float sa   = partials[N_BINS + i];
    const float sc   = partials[2 * N_BINS + i];
    const float prop = cnt * invN;
    const float den  = fmaxf(cnt, 1.0f);
    const float gap  = sc / den - sa / den;
    const bool  ne   = cnt > 0.0f;
    out[1 + i]          = ne ? gap * prop : 0.0f;           // bin_over_confidence
    out[1 + N_BINS + i] = prop;                             // prop_in_bin
    terms[i]            = ne ? fabsf(gap) * prop : 0.0f;
  }
  __syncthreads();
  if (i == 0) {
    float s = 0.f;
    for (int b = 0; b < N_BINS; ++b) s += terms[b];
    out[0] = s;                                             // ece
  }
}

// ---------------------------------------------------------------------------
extern "C" void kernel_launch(void* const* d_in, const int* in_sizes, int n_in,
                              void* d_out, int out_size, void* d_ws, size_t ws_size,
                              hipStream_t stream) {
  const float*     logits = (const float*)d_in[0];      // [N, 50] f32
  const long long* labels = (const long long*)d_in[1];  // [N] int64
  float*           out    = (float*)d_out;              // 31 floats
  float*           ws     = (float*)d_ws;

  const int n      = in_sizes[1];
  const int nTiles = (n + TILE_ROWS - 1) / TILE_ROWS;
  const int grid   = nTiles < MAX_GRID ? nTiles : MAX_GRID;

  (void)hipMemsetAsync(d_ws, 0, 3 * N_BINS * sizeof(float), stream);
  ece_partials_kernel<<<grid, BLOCK, 0, stream>>>(logits, labels, ws, n, nTiles);
  ece_finalize_kernel<<<1, 32, 0, stream>>>(ws, out, 1.0f / (float)n);
}